// ParallelRXComplex_30494267801966
// MI455X (gfx1250) — compile-verified
//
#include <hip/hip_runtime.h>

// MI455X / gfx1250: wave32, WMMA fp32 path.
// op = kron of 12 RX gates => three 16x16 complex group gates GA,GB,GC.
// One workgroup per row: load row (32KB) to LDS, 3 WMMA contraction stages,
// stream result to global. Bandwidth-bound: 256 MiB total -> ~11.5us roofline.

typedef float v2f __attribute__((ext_vector_type(2)));
typedef float v8f __attribute__((ext_vector_type(8)));

#define NQ    12
#define DIM   4096
#define NBATCH 4096

// ---------------- setup: build GA, GB, GC (16x16 complex each) ----------------
// Group g covers qubits 4g..4g+3; qubit (4g+t) maps to bit (3-t) of the 4-bit
// group index (qubit 0 is the MSB of the full index, per the kron order).
// RX gate: diag = cos(th/2), offdiag = -i sin(th/2).
// ws layout: g[grp*512 + jk] = real, g[grp*512 + 256 + jk] = imag (jk = j*16+k).
__global__ __launch_bounds__(256) void rx_build_gates(const float* __restrict__ params,
                                                      float* __restrict__ g) {
    int tid = blockIdx.x * 256 + threadIdx.x;
    if (tid >= 3 * 256) return;
    int grp = tid >> 8;
    int jk  = tid & 255;
    int j = jk >> 4, k = jk & 15;
    float pr = 1.0f, pi = 0.0f;
#pragma unroll
    for (int t = 0; t < 4; ++t) {
        int q = grp * 4 + t;
        float h = 0.5f * params[q];
        float c = cosf(h), s = sinf(h);
        int bj = (j >> (3 - t)) & 1;
        int bk = (k >> (3 - t)) & 1;
        float gr = (bj == bk) ? c : 0.0f;
        float gi = (bj == bk) ? 0.0f : -s;
        float nr = pr * gr - pi * gi;
        float ni = pr * gi + pi * gr;
        pr = nr; pi = ni;
    }
    g[grp * 512 + jk]       = pr;
    g[grp * 512 + 256 + jk] = pi;
}

// ---------------- WMMA fragment helpers (f32 16x16x4, wave32) ----------------
// A (16x4): lanes 0-15 hold M=lane, K = k0,k0+1 in vgpr .x/.y; lanes 16-31 K+=2.
// B (4x16): lanes 0-15 hold K = k0,k0+1 rows; lanes 16-31 K+=2; N = lane&15.
// C/D (16x16): vgpr v holds M = v + 8*(lane>=16), N = lane&15.
__device__ __forceinline__ v2f frag_a(const float* src, int stride, int k0, int r16) {
    v2f a;
    a.x = src[r16 * stride + k0];
    a.y = src[r16 * stride + k0 + 1];
    return a;
}
__device__ __forceinline__ v2f frag_b(const float* src, int stride, int k0, int r16) {
    v2f b;
    b.x = src[(k0)     * stride + r16];
    b.y = src[(k0 + 1) * stride + r16];
    return b;
}
__device__ __forceinline__ void store_d(float* dst, int stride, const v8f& d, int r16, int hi) {
#pragma unroll
    for (int v = 0; v < 8; ++v)
        dst[(v + 8 * hi) * stride + r16] = d[v];
}

// Complex 16x16x16 tile GEMM: (Dr + iDi) = (Ar + iAi) @ (Br + iBi)
// K=16 via 4 chained v_wmma_f32_16x16x4_f32 per real product (16 WMMAs total).
__device__ __forceinline__ void ctile(const float* Ar, const float* Ai, int sa,
                                      const float* Br, const float* Bi, int sb,
                                      v8f& dr, v8f& di, int r16, int hi) {
    v8f z = {0.f, 0.f, 0.f, 0.f, 0.f, 0.f, 0.f, 0.f};
    dr = z; di = z;
#pragma unroll
    for (int kb = 0; kb < 4; ++kb) {
        int k0 = kb * 4 + 2 * hi;
        v2f a_r = frag_a(Ar, sa, k0, r16);
        v2f a_i = frag_a(Ai, sa, k0, r16);
        v2f b_r = frag_b(Br, sb, k0, r16);
        v2f b_i = frag_b(Bi, sb, k0, r16);
        v2f nb_i; nb_i.x = -b_i.x; nb_i.y = -b_i.y;
        // Dr += Ar*Br - Ai*Bi ; Di += Ar*Bi + Ai*Br
        dr = __builtin_amdgcn_wmma_f32_16x16x4_f32(false, a_r, false, b_r,  (short)0, dr, false, false);
        dr = __builtin_amdgcn_wmma_f32_16x16x4_f32(false, a_i, false, nb_i, (short)0, dr, false, false);
        di = __builtin_amdgcn_wmma_f32_16x16x4_f32(false, a_r, false, b_i,  (short)0, di, false, false);
        di = __builtin_amdgcn_wmma_f32_16x16x4_f32(false, a_i, false, b_r,  (short)0, di, false, false);
    }
}

// ---------------- main kernel: one workgroup (8 waves) per row ----------------
__global__ __launch_bounds__(256) void rx_apply(const float* __restrict__ xr,
                                                const float* __restrict__ xi,
                                                const float* __restrict__ g,
                                                float* __restrict__ outr,
                                                float* __restrict__ outi) {
    __shared__ __align__(16) float sr[DIM];
    __shared__ __align__(16) float si[DIM];

    const int row = blockIdx.x;
    const size_t rbase = (size_t)row * DIM;

    // Cooperative row load: 4096 floats per buffer = 1024 float4; 256 threads x4.
    {
        const float4* xr4 = (const float4*)(xr + rbase);
        const float4* xi4 = (const float4*)(xi + rbase);
        float4* sr4 = (float4*)sr;
        float4* si4 = (float4*)si;
#pragma unroll
        for (int it = 0; it < 4; ++it) {
            int idx = threadIdx.x + it * 256;
            sr4[idx] = xr4[idx];
            si4[idx] = xi4[idx];
        }
    }
    __syncthreads();

    const int lane = threadIdx.x & 31;
    const int wave = threadIdx.x >> 5;
    const int r16  = lane & 15;
    const int hi   = lane >> 4;

    const float* gAr = g;        const float* gAi = g + 256;
    const float* gBr = g + 512;  const float* gBi = g + 768;
    const float* gCr = g + 1024; const float* gCi = g + 1280;

    v8f dr, di;

    // Stages 1+2. Wave w owns regions {w, w+8}; both stages stay inside the
    // wave's own region, so no barrier is needed between them (per-wave LDS
    // ordering guarantees read-before-overwrite for the in-place update).
#pragma unroll
    for (int rep = 0; rep < 2; ++rep) {
        const int t = wave + 8 * rep;
        float* br_ = sr + t * 256;
        float* bi_ = si + t * 256;
        // Stage 1: contract c-axis.  D[i][n] = sum_k X[t][i][k] * GC[k][n]
        ctile(br_, bi_, 16, gCr, gCi, 16, dr, di, r16, hi);
        store_d(br_, 16, dr, r16, hi);
        store_d(bi_, 16, di, r16, hi);
        // Stage 2: contract b-axis (GB symmetric -> left-multiply).
        // D[i][n] = sum_k GB[i][k] * T1[t][k][n]
        ctile(gBr, gBi, 16, br_, bi_, 16, dr, di, r16, hi);
        store_d(br_, 16, dr, r16, hi);
        store_d(bi_, 16, di, r16, hi);
    }

    __syncthreads();  // stage 3 reads across all regions

    // Stage 3: contract a-axis (GA symmetric). View buffer as 16x256 matrix,
    // column block t: D[i][n] = sum_k GA[i][k] * T2[k][16t+n]; store to global.
#pragma unroll
    for (int rep = 0; rep < 2; ++rep) {
        const int t = wave + 8 * rep;
        ctile(gAr, gAi, 16, sr + 16 * t, si + 16 * t, 256, dr, di, r16, hi);
#pragma unroll
        for (int v = 0; v < 8; ++v) {
            size_t o = rbase + (size_t)((v + 8 * hi) * 256 + 16 * t + r16);
            outr[o] = dr[v];
            outi[o] = di[v];
        }
    }
}

// ---------------- launch ----------------
extern "C" void kernel_launch(void* const* d_in, const int* in_sizes, int n_in,
                              void* d_out, int out_size, void* d_ws, size_t ws_size,
                              hipStream_t stream) {
    (void)in_sizes; (void)n_in; (void)out_size; (void)ws_size;
    const float* xr     = (const float*)d_in[0];
    const float* xi     = (const float*)d_in[1];
    const float* params = (const float*)d_in[2];

    float* g = (float*)d_ws;  // 3 groups * (256 real + 256 imag) = 1536 floats
    float* outr = (float*)d_out;
    float* outi = outr + (size_t)NBATCH * DIM;

    rx_build_gates<<<3, 256, 0, stream>>>(params, g);
    rx_apply<<<NBATCH, 256, 0, stream>>>(xr, xi, g, outr, outi);
}